// SpiralEnblock_2843268350430
// MI455X (gfx1250) — compile-verified
//
#include <hip/hip_runtime.h>
#include <cstdint>

typedef __attribute__((ext_vector_type(2))) float v2f;
typedef __attribute__((ext_vector_type(8))) float v8f;

#define B_SZ   8
#define N_VERT 50000
#define CH_IN  32
#define L_SP   9
#define CH_OUT 64
#define M_OUT  12500
#define NNZ_E  37500
#define K_TOT  (L_SP * CH_IN)   // 288
#define ETILE  32               // NNZ entries per block (2 WMMA m-tiles)

// ---------------------------------------------------------------------------
// Zero the pooled output (harness poisons d_out before timing).
// ---------------------------------------------------------------------------
__global__ void spiral_zero_kernel(float* __restrict__ out, int n) {
    int i = blockIdx.x * blockDim.x + threadIdx.x;
    if (i < n) out[i] = 0.0f;
}

// ---------------------------------------------------------------------------
// Fused: spiral-gather -> (32x288)x(288x64) WMMA GEMM + bias -> ELU ->
//        value-scale -> atomic scatter-add into pooled[b, rows, :].
// Block = 32 NNZ entries, 8 waves; wave w == batch b (B==8).
// Each wave: two 16-row m-tiles so every B fragment feeds 2 WMMAs.
// ---------------------------------------------------------------------------
__launch_bounds__(256)
__global__ void spiral_fused_kernel(const float* __restrict__ x,
                                    const float* __restrict__ weight,
                                    const float* __restrict__ bias,
                                    const float* __restrict__ values,
                                    const int*   __restrict__ spiral_indices,
                                    const int*   __restrict__ rows,
                                    const int*   __restrict__ cols,
                                    float* __restrict__ out) {
    // Weight staged K-pair-interleaved: sW2[(k>>1)*64 + ch] = {W[k][ch], W[k+1][ch]}
    // so each lane's B fragment is a single contiguous ds_load_b64.
    __shared__ float sW2[K_TOT * CH_OUT];      // 72 KB
    __shared__ int   sIdx[ETILE * L_SP];       // spiral indices of gathered verts
    __shared__ int   sRows[ETILE];
    __shared__ float sVals[ETILE];
    __shared__ int   sCols[ETILE];

    const int tid = threadIdx.x;
    const int e0  = blockIdx.x * ETILE;

    // --- stage entry metadata (pad invalid entries with value=0 => no-op adds)
    if (tid < ETILE) {
        int e = e0 + tid;
        if (e < NNZ_E) {
            sCols[tid] = cols[e];
            sRows[tid] = rows[e];
            sVals[tid] = values[e];
        } else {
            sCols[tid] = 0; sRows[tid] = 0; sVals[tid] = 0.0f;
        }
    }
    // --- stage weight: coalesced float4 global reads, K-pair-interleaved stores
    {
        const float4* w4 = (const float4*)weight;
        #pragma unroll
        for (int i = 0; i < (K_TOT * CH_OUT / 4) / 256; ++i) {   // 18 iters
            int idx4 = tid + i * 256;
            int k    = idx4 >> 4;            // 16 float4 per 64-wide row
            int ch0  = (idx4 & 15) * 4;
            float4 w = w4[idx4];
            float* dst = sW2 + (k >> 1) * (CH_OUT * 2) + (k & 1);
            dst[(ch0 + 0) * 2] = w.x;
            dst[(ch0 + 1) * 2] = w.y;
            dst[(ch0 + 2) * 2] = w.z;
            dst[(ch0 + 3) * 2] = w.w;
        }
    }
    __syncthreads();
    // --- stage spiral indices for the 32 gathered source vertices
    for (int t = tid; t < ETILE * L_SP; t += 256) {
        int i = t / L_SP, l = t - i * L_SP;
        sIdx[t] = spiral_indices[sCols[i] * L_SP + l];
    }
    __syncthreads();

    const int lane  = tid & 31;
    const int b     = tid >> 5;      // wave id == batch index
    const int laneN = lane & 15;
    const int half  = lane >> 4;

    // C/D layout: VGPR j -> M=j (lanes 0-15) / M=j+8 (lanes 16-31); N = lane&15.
    // acc[mt][t]: m-tile mt (entries mt*16..mt*16+15), channel tile t (ch t*16..).
    v8f acc[2][4];
    #pragma unroll
    for (int t = 0; t < 4; ++t) {
        float bb = bias[t * 16 + laneN];
        v8f bv = (v8f){bb, bb, bb, bb, bb, bb, bb, bb};
        acc[0][t] = bv;
        acc[1][t] = bv;
    }

    // A layout (16x4 f32): lane row m=laneN(+16*mt); VGPR0 K=2*half, VGPR1 K=2*half+1
    // => one contiguous float2 global load per K-step per m-tile.
    #pragma unroll
    for (int l = 0; l < L_SP; ++l) {
        const float* arow0 =
            x + ((size_t)b * N_VERT + (size_t)sIdx[laneN * L_SP + l]) * CH_IN;
        const float* arow1 =
            x + ((size_t)b * N_VERT + (size_t)sIdx[(laneN + 16) * L_SP + l]) * CH_IN;
        #pragma unroll
        for (int kc = 0; kc < CH_IN / 4; ++kc) {
            const int k0 = kc * 4;
            v2f a0 = *(const v2f*)(arow0 + k0 + 2 * half);
            v2f a1 = *(const v2f*)(arow1 + k0 + 2 * half);
            // B pair index: kg = l*32 + k0 + 2*half (even); kg>>1 = l*16 + 2*kc + half
            const int kp = l * 16 + kc * 2 + half;
            #pragma unroll
            for (int t = 0; t < 4; ++t) {
                v2f bf = *(const v2f*)(sW2 + (kp * CH_OUT + t * 16 + laneN) * 2);
                acc[0][t] = __builtin_amdgcn_wmma_f32_16x16x4_f32(
                    false, a0, false, bf, (short)0, acc[0][t], false, false);
                acc[1][t] = __builtin_amdgcn_wmma_f32_16x16x4_f32(
                    false, a1, false, bf, (short)0, acc[1][t], false, false);
            }
        }
    }

    // --- epilogue: ELU, scale by edge value, atomic scatter-add
    #pragma unroll
    for (int mt = 0; mt < 2; ++mt) {
        #pragma unroll
        for (int t = 0; t < 4; ++t) {
            const int ch = t * 16 + laneN;
            #pragma unroll
            for (int j = 0; j < 8; ++j) {
                const int m = mt * 16 + j + 8 * half;
                float v = acc[mt][t][j];
                v = (v > 0.0f) ? v : expm1f(v);      // jax.nn.elu, alpha=1
                v *= sVals[m];
                atomicAdd(out + ((size_t)b * M_OUT + (size_t)sRows[m]) * CH_OUT + ch, v);
            }
        }
    }
}

// ---------------------------------------------------------------------------
extern "C" void kernel_launch(void* const* d_in, const int* in_sizes, int n_in,
                              void* d_out, int out_size, void* d_ws, size_t ws_size,
                              hipStream_t stream) {
    const float* x      = (const float*)d_in[0];
    const float* weight = (const float*)d_in[1];
    const float* bias   = (const float*)d_in[2];
    const float* values = (const float*)d_in[3];
    const int*   spiral = (const int*)d_in[4];
    const int*   rows   = (const int*)d_in[5];
    const int*   cols   = (const int*)d_in[6];
    float* out = (float*)d_out;

    spiral_zero_kernel<<<(out_size + 255) / 256, 256, 0, stream>>>(out, out_size);

    const int blocks = (NNZ_E + ETILE - 1) / ETILE;  // 1172 entry-tiles
    spiral_fused_kernel<<<blocks, 256, 0, stream>>>(
        x, weight, bias, values, spiral, rows, cols, out);
}